// GAT2_28372553957492
// MI455X (gfx1250) — compile-verified
//
#include <hip/hip_runtime.h>
#include <math.h>

// ---------------------------------------------------------------------------
// GAT (3x GATConv) for MI455X / gfx1250, wave32.
//  - GEMMs: V_WMMA_F32_16X16X4_F32 (full fp32; workload is L2-bound, so no
//    payoff for reduced precision).
//  - Edge softmax: 3 passes with L2 atomics; per-(edge,head) logits / exp
//    values cached in a workspace buffer to avoid recomputation.
//  - Aggregation: one wave per (edge,head), lane = feature element ->
//    coalesced 128B gathers and contiguous atomic bursts.
// ---------------------------------------------------------------------------

typedef __attribute__((ext_vector_type(2))) float v2f;
typedef __attribute__((ext_vector_type(8))) float v8f;

#define HEADS 4
#define LRELU_SLOPE 0.2f
#define NEG_INF_KEY 0x007FFFFFu   // ordered-key encoding of -inf

// Monotonic float -> uint mapping (IEEE total order for non-NaN values)
__device__ __forceinline__ unsigned f2ord(float f) {
    unsigned u = __float_as_uint(f);
    return (u & 0x80000000u) ? ~u : (u | 0x80000000u);
}
__device__ __forceinline__ float ord2f(unsigned u) {
    unsigned b = (u & 0x80000000u) ? (u & 0x7FFFFFFFu) : ~u;
    return __uint_as_float(b);
}
__device__ __forceinline__ float lrelu(float v) {
    return v > 0.0f ? v : LRELU_SLOPE * v;
}

// ---------------------------------------------------------------------------
// fp32 WMMA GEMM: C[M,F] = A[M,K] @ B[K,F].  One 16x16 output tile per wave,
// K stepped by 4 through v_wmma_f32_16x16x4_f32.  M, F multiples of 16; K of 4.
// ---------------------------------------------------------------------------
__global__ void k_gemm_f32_wmma(const float* __restrict__ A,
                                const float* __restrict__ B,
                                float* __restrict__ C,
                                int M, int K, int F) {
    const int wavesPerBlock = blockDim.x >> 5;
    const int wave = blockIdx.x * wavesPerBlock + (threadIdx.x >> 5);
    const int nf = F >> 4;
    const int mtiles = M >> 4;
    if (wave >= mtiles * nf) return;               // wave-uniform: EXEC stays all-1s
    const int tm = wave / nf;
    const int tn = wave % nf;
    const int lane = threadIdx.x & 31;
    const int half = lane >> 4;                    // 0: K pair {0,1}; 1: K pair {2,3}
    const int l16  = lane & 15;

    const float* arow = A + (size_t)(tm * 16 + l16) * K;          // A-row for this lane
    const float* bcol = B + (size_t)(tn * 16 + l16);              // B-col for this lane

    v8f acc = {};
    for (int k0 = 0; k0 < K; k0 += 4) {
        const int ka = k0 + 2 * half;
        v2f a, b;
        a.x = arow[ka];
        a.y = arow[ka + 1];
        b.x = bcol[(size_t)ka * F];
        b.y = bcol[(size_t)(ka + 1) * F];
        acc = __builtin_amdgcn_wmma_f32_16x16x4_f32(
            /*neg_a=*/false, a, /*neg_b=*/false, b,
            /*c_mod=*/(short)0, acc, /*reuse_a=*/false, /*reuse_b=*/false);
    }
    // C/D layout: VGPR j -> row M = j (lanes 0-15) / M = 8+j (lanes 16-31)
    float* crow = C + (size_t)(tm * 16 + half * 8) * F + tn * 16 + l16;
#pragma unroll
    for (int j = 0; j < 8; ++j) crow[(size_t)j * F] = acc[j];
}

// ---------------------------------------------------------------------------
// el[n,h] = <feat[n,h,:], attn_l[h,:]>, er likewise.  One thread per (n,h).
// ---------------------------------------------------------------------------
__global__ void k_attn_dots(const float* __restrict__ feat,
                            const float* __restrict__ al,
                            const float* __restrict__ ar,
                            float* __restrict__ el, float* __restrict__ er,
                            int N, int D) {
    int i = blockIdx.x * blockDim.x + threadIdx.x;
    if (i >= N * HEADS) return;
    int n = i / HEADS, h = i % HEADS;
    const float* f = feat + (size_t)n * HEADS * D + h * D;
    const float* a = al + h * D;
    const float* b = ar + h * D;
    float sl = 0.f, sr = 0.f;
    for (int d = 0; d < D; ++d) { float v = f[d]; sl += v * a[d]; sr += v * b[d]; }
    el[i] = sl;
    er[i] = sr;
}

__global__ void k_init_f32(float* __restrict__ p, int n, float v) {
    int i = blockIdx.x * blockDim.x + threadIdx.x;
    if (i < n) p[i] = v;
}
__global__ void k_init_u32(unsigned* __restrict__ p, int n, unsigned v) {
    int i = blockIdx.x * blockDim.x + threadIdx.x;
    if (i < n) p[i] = v;
}

// Pass A: compute leaky-relu logit per (edge,head), cache it, and
// segment-max over dst via atomicMax on ordered uint keys (L2 atomics).
__global__ void k_edge_max(const int* __restrict__ src, const int* __restrict__ dst,
                           const float* __restrict__ el, const float* __restrict__ er,
                           unsigned* __restrict__ nkey, float* __restrict__ eval,
                           int E) {
    int i = blockIdx.x * blockDim.x + threadIdx.x;
    if (i >= E * HEADS) return;
    int e = i / HEADS, h = i % HEADS;
    int s = src[e], d = dst[e];
    float v = lrelu(el[s * HEADS + h] + er[d * HEADS + h]);
    eval[i] = v;
    atomicMax(&nkey[d * HEADS + h], f2ord(v));
}

__global__ void k_decode_max(const unsigned* __restrict__ nkey,
                             float* __restrict__ nmax, int n) {
    int i = blockIdx.x * blockDim.x + threadIdx.x;
    if (i >= n) return;
    unsigned u = nkey[i];
    nmax[i] = (u == NEG_INF_KEY) ? 0.0f : ord2f(u);   // zero-in-degree -> 0
}

// Pass B: ex = exp(e - max); overwrite cache with ex; denom[dst,h] += ex
__global__ void k_edge_sum(const int* __restrict__ dst,
                           const float* __restrict__ nmax,
                           float* __restrict__ nsum, float* __restrict__ eval,
                           int E) {
    int i = blockIdx.x * blockDim.x + threadIdx.x;
    if (i >= E * HEADS) return;
    int e = i / HEADS, h = i % HEADS;
    int dh = dst[e] * HEADS + h;
    float ex = __expf(eval[i] - nmax[dh]);
    eval[i] = ex;
    atomicAdd(&nsum[dh], ex);
}

// Pass C: one WAVE per (edge,head); lane = feature element.
// feat[src] gather and out[dst] atomics are contiguous per wave.
__global__ void k_edge_aggr(const int* __restrict__ src, const int* __restrict__ dst,
                            const float* __restrict__ nsum,
                            const float* __restrict__ eval,
                            const float* __restrict__ feat, float* __restrict__ out,
                            int E, int D) {
    const int wave = blockIdx.x * (blockDim.x >> 5) + (threadIdx.x >> 5);
    if (wave >= E * HEADS) return;
    const int lane = threadIdx.x & 31;
    const int e = wave / HEADS, h = wave % HEADS;
    const int s = src[e], d = dst[e];          // same-address loads -> broadcast
    const float alpha = eval[wave] / fmaxf(nsum[d * HEADS + h], 1e-9f);
    const float* fs = feat + (size_t)s * HEADS * D + h * D;
    float* od = out + (size_t)d * HEADS * D + h * D;
    for (int dd = lane; dd < D; dd += 32)      // D=32: one coalesced 128B burst
        atomicAdd(&od[dd], alpha * fs[dd]);
}

// Between layers: h_next[n,d] = mean_h relu(agg[n,h,d] + bias[h,d])
__global__ void k_merge_heads(const float* __restrict__ agg, const float* __restrict__ bias,
                              float* __restrict__ hout, int N, int D) {
    int i = blockIdx.x * blockDim.x + threadIdx.x;
    if (i >= N * D) return;
    int n = i / D, d = i % D;
    float s = 0.f;
#pragma unroll
    for (int h = 0; h < HEADS; ++h) {
        float v = agg[(size_t)n * HEADS * D + h * D + d] + bias[h * D + d];
        s += v > 0.f ? v : 0.f;
    }
    hout[i] = s * (1.0f / HEADS);
}

__global__ void k_add_bias(float* __restrict__ out, const float* __restrict__ b,
                           int total, int F) {
    int i = blockIdx.x * blockDim.x + threadIdx.x;
    if (i < total) out[i] += b[i % F];
}

// ---------------------------------------------------------------------------
static inline int cdiv(int a, int b) { return (a + b - 1) / b; }

static void run_gat_layer(const float* hin, int K,
                          const float* W, const float* al, const float* ar,
                          int D, const int* src, const int* dst, int N, int E,
                          float* feat, float* el, float* er, float* nmax,
                          float* nsum, unsigned* nkey, float* eval,
                          float* aggOut, hipStream_t stream) {
    const int F = HEADS * D;
    const int tiles = (N / 16) * (F / 16);
    const int NH = N * HEADS;
    const int EH = E * HEADS;

    k_gemm_f32_wmma<<<cdiv(tiles, 8), 256, 0, stream>>>(hin, W, feat, N, K, F);
    k_attn_dots<<<cdiv(NH, 256), 256, 0, stream>>>(feat, al, ar, el, er, N, D);
    k_init_u32<<<cdiv(NH, 256), 256, 0, stream>>>(nkey, NH, NEG_INF_KEY);
    k_init_f32<<<cdiv(NH, 256), 256, 0, stream>>>(nsum, NH, 0.0f);
    k_init_f32<<<cdiv(N * F, 256), 256, 0, stream>>>(aggOut, N * F, 0.0f);
    k_edge_max<<<cdiv(EH, 256), 256, 0, stream>>>(src, dst, el, er, nkey, eval, E);
    k_decode_max<<<cdiv(NH, 256), 256, 0, stream>>>(nkey, nmax, NH);
    k_edge_sum<<<cdiv(EH, 256), 256, 0, stream>>>(dst, nmax, nsum, eval, E);
    // one wave per (edge,head): EH waves, 8 waves per 256-thread block
    k_edge_aggr<<<cdiv(EH, 8), 256, 0, stream>>>(src, dst, nsum, eval, feat,
                                                 aggOut, E, D);
}

extern "C" void kernel_launch(void* const* d_in, const int* in_sizes, int n_in,
                              void* d_out, int out_size, void* d_ws, size_t ws_size,
                              hipStream_t stream) {
    const float* x   = (const float*)d_in[0];
    const int*   src = (const int*)d_in[1];
    const int*   dst = (const int*)d_in[2];
    const float* W0  = (const float*)d_in[3];
    const float* al0 = (const float*)d_in[4];
    const float* ar0 = (const float*)d_in[5];
    const float* b0  = (const float*)d_in[6];
    const float* W1  = (const float*)d_in[7];
    const float* al1 = (const float*)d_in[8];
    const float* ar1 = (const float*)d_in[9];
    const float* b1  = (const float*)d_in[10];
    const float* W2  = (const float*)d_in[11];
    const float* al2 = (const float*)d_in[12];
    const float* ar2 = (const float*)d_in[13];
    const float* b2  = (const float*)d_in[14];

    const int IN = 128, HID = 32, CLS = 40;
    const int N = in_sizes[0] / IN;    // 50000
    const int E = in_sizes[1];         // 800000

    // Workspace carve-up (floats): ~81 MB total
    float*    feat = (float*)d_ws;                          // N * 160 (max F)
    float*    el   = feat + (size_t)N * HEADS * CLS;        // N*H
    float*    er   = el   + (size_t)N * HEADS;              // N*H
    float*    nmax = er   + (size_t)N * HEADS;              // N*H
    float*    nsum = nmax + (size_t)N * HEADS;              // N*H
    unsigned* nkey = (unsigned*)(nsum + (size_t)N * HEADS); // N*H
    float*    eval = (float*)nkey + (size_t)N * HEADS;      // E*H
    float*    agg  = eval + (size_t)E * HEADS;              // N * 128
    float*    hbuf = agg + (size_t)N * HEADS * HID;         // N * HID

    float* out = (float*)d_out;                             // [N, HEADS*CLS]

    // Layer 0: IN -> H x HID
    run_gat_layer(x, IN, W0, al0, ar0, HID, src, dst, N, E,
                  feat, el, er, nmax, nsum, nkey, eval, agg, stream);
    k_merge_heads<<<cdiv(N * HID, 256), 256, 0, stream>>>(agg, b0, hbuf, N, HID);

    // Layer 1: HID -> H x HID
    run_gat_layer(hbuf, HID, W1, al1, ar1, HID, src, dst, N, E,
                  feat, el, er, nmax, nsum, nkey, eval, agg, stream);
    k_merge_heads<<<cdiv(N * HID, 256), 256, 0, stream>>>(agg, b1, hbuf, N, HID);

    // Layer 2: HID -> H x CLS, aggregate straight into d_out, then add bias
    run_gat_layer(hbuf, HID, W2, al2, ar2, CLS, src, dst, N, E,
                  feat, el, er, nmax, nsum, nkey, eval, out, stream);
    k_add_bias<<<cdiv(N * HEADS * CLS, 256), 256, 0, stream>>>(out, b2,
                                                               N * HEADS * CLS, HEADS * CLS);
}